// HierarchicalSpatialAnalyzer_27333171871929
// MI455X (gfx1250) — compile-verified
//
#include <hip/hip_runtime.h>
#include <hip/hip_bf16.h>
#include <stdint.h>

#define S 64
#define NPF 147456            // 384*384
#define NE (S * NPF)          // 9437184
#define H1BINS 16384
#define H2BINS 65536
#define R0 73727u             // lower median rank (0-indexed)
#define R1 73728u             // upper median rank
#define TILE 32

typedef float v2f __attribute__((ext_vector_type(2)));
typedef float v8f __attribute__((ext_vector_type(8)));

#if defined(__has_builtin)
#if __has_builtin(__builtin_amdgcn_wmma_f32_16x16x4_f32)
#define HAVE_WMMA_F32 1
#endif
#if __has_builtin(__builtin_amdgcn_sqrtf)
#define FAST_SQRT(x) __builtin_amdgcn_sqrtf(x)
#endif
#endif
#ifndef HAVE_WMMA_F32
#define HAVE_WMMA_F32 0
#endif
#ifndef FAST_SQRT
#define FAST_SQRT(x) sqrtf(x)
#endif

// ---------------- workspace layout ----------------
#define WS_HIST1 0
#define WS_HIST2 (S * H1BINS * 4)              // 4 MiB in
#define WS_META  (WS_HIST2 + S * H2BINS * 4)   // 20 MiB in

struct Meta {
  int tb0[S]; int rw0[S];      // bin + rank-within for rank R0
  int tb1[S]; int rw1[S];      // bin + rank-within for rank R1
  unsigned minPat[S];          // min conf bit pattern in bin tb1 (if tb1!=tb0)
  float med[S];
  unsigned bboxU[6];           // 0..2 ordered-min, 3..5 ordered-max
  float pminf[3];
  float vox[4];                // voxel size per scale = min_extent / lambda
  unsigned pad;
  double sums[3];              // n, sum(d*m), sum(d^2*m)
};

__device__ __forceinline__ unsigned ford(float f) {
  unsigned u = __float_as_uint(f);
  return (u & 0x80000000u) ? ~u : (u | 0x80000000u);
}
__device__ __forceinline__ float funord(unsigned u) {
  unsigned b = (u & 0x80000000u) ? (u ^ 0x80000000u) : ~u;
  return __uint_as_float(b);
}

// ---------------- init ----------------
__global__ void k_init(unsigned* ws_u32, Meta* meta) {
  size_t i = (size_t)blockIdx.x * blockDim.x + threadIdx.x;
  size_t nwords = WS_META / 4;
  size_t stride = (size_t)gridDim.x * blockDim.x;
  for (size_t j = i; j < nwords; j += stride) ws_u32[j] = 0u;
  if (blockIdx.x == 0) {
    int t = threadIdx.x;
    if (t < S) meta->minPat[t] = 0xFFFFFFFFu;
    if (t >= 64 && t < 70) { int d = t - 64; meta->bboxU[d] = (d < 3) ? 0xFFFFFFFFu : 0u; }
    if (t == 70) { meta->sums[0] = 0.0; meta->sums[1] = 0.0; meta->sums[2] = 0.0; }
  }
}

// ---------------- median pass 1: top-16-bit histogram ----------------
__global__ __launch_bounds__(256) void k_hist1(const float* conf, unsigned* hist1) {
  size_t stride = (size_t)gridDim.x * blockDim.x;
  for (size_t i = (size_t)blockIdx.x * blockDim.x + threadIdx.x; i < NE; i += stride) {
    int f = (int)(i / NPF);
    unsigned b = __float_as_uint(conf[i]) >> 16;   // conf in [0,1) => b < 16384
    atomicAdd(&hist1[(size_t)f * H1BINS + b], 1u);
  }
}

__global__ __launch_bounds__(256) void k_findbin(const unsigned* hist1, Meta* meta) {
  int f = blockIdx.x;
  const unsigned* h = hist1 + (size_t)f * H1BINS;
  __shared__ unsigned part[256]; __shared__ unsigned base[256];
  int t = threadIdx.x;
  unsigned s = 0;
  for (int i = 0; i < 64; ++i) s += h[t * 64 + i];
  part[t] = s; __syncthreads();
  if (t == 0) { unsigned a = 0; for (int i = 0; i < 256; ++i) { base[i] = a; a += part[i]; } }
  __syncthreads();
  unsigned a = base[t];
  for (int i = 0; i < 64; ++i) {
    unsigned c0 = a; a += h[t * 64 + i];
    if (R0 >= c0 && R0 < a) { meta->tb0[f] = t * 64 + i; meta->rw0[f] = (int)(R0 - c0); }
    if (R1 >= c0 && R1 < a) { meta->tb1[f] = t * 64 + i; meta->rw1[f] = (int)(R1 - c0); }
  }
}

// ---------------- median pass 2: low-16-bit refinement ----------------
__global__ __launch_bounds__(256) void k_hist2(const float* conf, const Meta* meta,
                                               unsigned* hist2, Meta* metaw) {
  size_t stride = (size_t)gridDim.x * blockDim.x;
  for (size_t i = (size_t)blockIdx.x * blockDim.x + threadIdx.x; i < NE; i += stride) {
    int f = (int)(i / NPF);
    unsigned b = __float_as_uint(conf[i]);
    int top = (int)(b >> 16);
    if (top == meta->tb0[f]) {
      atomicAdd(&hist2[(size_t)f * H2BINS + (b & 0xFFFFu)], 1u);
    } else if (top == meta->tb1[f]) {
      atomicMin(&metaw->minPat[f], b);   // smallest value in upper bin
    }
  }
}

__global__ __launch_bounds__(256) void k_findmed(const unsigned* hist2, Meta* meta) {
  int f = blockIdx.x;
  const unsigned* h = hist2 + (size_t)f * H2BINS;
  __shared__ unsigned part[256]; __shared__ unsigned base[256];
  __shared__ unsigned low0, low1;
  int t = threadIdx.x;
  unsigned s = 0;
  for (int i = 0; i < 256; ++i) s += h[t * 256 + i];
  part[t] = s; __syncthreads();
  if (t == 0) { unsigned a = 0; for (int i = 0; i < 256; ++i) { base[i] = a; a += part[i]; } }
  __syncthreads();
  bool same = (meta->tb1[f] == meta->tb0[f]);
  unsigned want0 = (unsigned)meta->rw0[f];
  unsigned want1 = same ? (unsigned)meta->rw1[f] : 0xFFFFFFFFu;
  unsigned a = base[t];
  for (int i = 0; i < 256; ++i) {
    unsigned c0 = a; a += h[t * 256 + i];
    if (want0 >= c0 && want0 < a) low0 = (unsigned)(t * 256 + i);
    if (want1 != 0xFFFFFFFFu && want1 >= c0 && want1 < a) low1 = (unsigned)(t * 256 + i);
  }
  __syncthreads();
  if (t == 0) {
    unsigned hi = (unsigned)meta->tb0[f] << 16;
    float v0 = __uint_as_float(hi | low0);
    float v1 = same ? __uint_as_float(hi | low1) : __uint_as_float(meta->minPat[f]);
    meta->med[f] = 0.5f * (v0 + v1);
  }
}

// ---------------- global bbox over valid points ----------------
__global__ __launch_bounds__(256) void k_bbox(const float* pts, const float* conf, Meta* meta) {
  __shared__ unsigned sb[6];
  int t = threadIdx.x;
  if (t < 6) sb[t] = (t < 3) ? 0xFFFFFFFFu : 0u;
  __syncthreads();
  unsigned lmin[3] = {0xFFFFFFFFu, 0xFFFFFFFFu, 0xFFFFFFFFu};
  unsigned lmax[3] = {0u, 0u, 0u};
  size_t stride = (size_t)gridDim.x * blockDim.x;
  for (size_t i = (size_t)blockIdx.x * blockDim.x + t; i < NE; i += stride) {
    int f = (int)(i / NPF);
    float c = conf[i];
    if (c > 0.1f && c >= meta->med[f]) {
      const float* p = pts + i * 3;
      for (int d = 0; d < 3; ++d) {
        unsigned u = ford(p[d]);
        lmin[d] = lmin[d] < u ? lmin[d] : u;
        lmax[d] = lmax[d] > u ? lmax[d] : u;
      }
    }
  }
  for (int d = 0; d < 3; ++d) { atomicMin(&sb[d], lmin[d]); atomicMax(&sb[3 + d], lmax[d]); }
  __syncthreads();
  if (t < 3) atomicMin(&meta->bboxU[t], sb[t]);
  else if (t < 6) atomicMax(&meta->bboxU[t], sb[t]);
}

__global__ void k_finbbox(Meta* meta) {
  float pmin[3], pmax[3];
  for (int d = 0; d < 3; ++d) {
    pmin[d] = funord(meta->bboxU[d]);
    pmax[d] = funord(meta->bboxU[3 + d]);
    meta->pminf[d] = pmin[d];
  }
  float ext = fminf(fminf(pmax[0] - pmin[0], pmax[1] - pmin[1]), pmax[2] - pmin[2]);
  const float lam[4] = {10.f, 20.f, 40.f, 20.f};
  for (int s = 0; s < 4; ++s) meta->vox[s] = ext / lam[s];
}

// ---------------- async LDS staging helper ----------------
__device__ __forceinline__ void stage_tile(const float* conf, const float* pts,
                                           size_t ebase, float* dst /*[4][TILE]*/, int L) {
  unsigned long long ac = (unsigned long long)(conf + ebase + (size_t)L);
  unsigned long long ap = (unsigned long long)(pts + (ebase + (size_t)L) * 3);
  unsigned dc = (unsigned)(unsigned long long)(dst + 0 * TILE + L);
  unsigned dx = (unsigned)(unsigned long long)(dst + 1 * TILE + L);
  unsigned dy = (unsigned)(unsigned long long)(dst + 2 * TILE + L);
  unsigned dz = (unsigned)(unsigned long long)(dst + 3 * TILE + L);
  asm volatile("global_load_async_to_lds_b32 %0, %1, off" :: "v"(dc), "v"(ac) : "memory");
  asm volatile("global_load_async_to_lds_b32 %0, %1, off" :: "v"(dx), "v"(ap) : "memory");
  asm volatile("global_load_async_to_lds_b32 %0, %1, off" :: "v"(dy), "v"(ap + 4ull) : "memory");
  asm volatile("global_load_async_to_lds_b32 %0, %1, off" :: "v"(dz), "v"(ap + 8ull) : "memory");
}

// ---------------- unique voxel counts: 1 block per (scale, frame) ----------------
__global__ __launch_bounds__(256) void k_voxel(const float* pts, const float* conf,
                                               const Meta* meta, float* out) {
  __shared__ unsigned bitmap[8192];          // 64^3 bits = 32 KB
  __shared__ float stg[8][2][4][TILE];       // per-wave double buffer: conf,x,y,z
  __shared__ int scnt;
  const int s = blockIdx.x >> 6;             // scale index 0..3
  const int f = blockIdx.x & 63;             // frame index 0..63
  const float medv = meta->med[f];
  const float p0 = meta->pminf[0], p1 = meta->pminf[1], p2 = meta->pminf[2];
  const float vox = meta->vox[s];
  const int wave = threadIdx.x >> 5;
  const int L = threadIdx.x & 31;
  const size_t fbase = (size_t)f * NPF;
  const int tilesPerWave = (NPF / TILE) / 8;     // 576

  for (int i = threadIdx.x; i < 8192; i += 256) bitmap[i] = 0u;
  if (threadIdx.x == 0) scnt = 0;
  __syncthreads();

  // prologue: stage first tile into buffer 0
  stage_tile(conf, pts, fbase + (size_t)wave * TILE, &stg[wave][0][0][0], L);
  for (int k = 0; k < tilesPerWave; ++k) {
    if (k + 1 < tilesPerWave) {
      int tn = wave + (k + 1) * 8;
      stage_tile(conf, pts, fbase + (size_t)tn * TILE, &stg[wave][(k + 1) & 1][0][0], L);
      asm volatile("s_wait_asynccnt 0x4" ::: "memory");   // previous tile's 4 loads done
    } else {
      asm volatile("s_wait_asynccnt 0x0" ::: "memory");
    }
    const float* buf = &stg[wave][k & 1][0][0];
    float c = buf[0 * TILE + L];
    float x = buf[1 * TILE + L];
    float y = buf[2 * TILE + L];
    float z = buf[3 * TILE + L];
    if (c > 0.1f && c >= medv) {
      int ix = (int)floorf((x - p0) / vox); ix = ix < 0 ? 0 : (ix > 63 ? 63 : ix);
      int iy = (int)floorf((y - p1) / vox); iy = iy < 0 ? 0 : (iy > 63 ? 63 : iy);
      int iz = (int)floorf((z - p2) / vox); iz = iz < 0 ? 0 : (iz > 63 ? 63 : iz);
      int bit = (ix << 12) | (iy << 6) | iz;
      atomicOr(&bitmap[bit >> 5], 1u << (bit & 31));
    }
  }
  __syncthreads();
  int local = 0;
  for (int i = threadIdx.x; i < 8192; i += 256) local += __popc(bitmap[i]);
  atomicAdd(&scnt, local);
  __syncthreads();
  if (threadIdx.x == 0) out[blockIdx.x] = (float)scnt;   // out[s*S + f]
}

// ---------------- complexity: dot products on the WMMA pipe ----------------
// For V_WMMA_F32_16X16X4_F32, lane L supplies A(M=L&15, K=(L>>4)*2+v),
// i.e. elements e = (L&15)*4 + (L>>4)*2 + v (v=0,1) -> two consecutive elements.
// With B mirrored the same way, diag(D) accumulates chunked dot products.
__global__ __launch_bounds__(256) void k_complex(const float* pts, const float* conf, Meta* meta) {
  const int gtid = blockIdx.x * blockDim.x + threadIdx.x;
  const int wave = gtid >> 5;
  const int nwaves = (gridDim.x * blockDim.x) >> 5;
  const int L = threadIdx.x & 31;
  const int eoff = ((L & 15) << 2) | ((L >> 4) << 1);   // even
  v8f c1 = {0.f,0.f,0.f,0.f,0.f,0.f,0.f,0.f};  // sum m
  v8f c2 = {0.f,0.f,0.f,0.f,0.f,0.f,0.f,0.f};  // sum d*m
  v8f c3 = {0.f,0.f,0.f,0.f,0.f,0.f,0.f,0.f};  // sum d^2*m
  const int NCH = NE / 64;                     // 147456 chunks of 64
  for (int ch = wave; ch < NCH; ch += nwaves) {
    int e0 = ch * 64 + eoff;                   // even -> 8B-aligned f32 pairs
    float2 cc = *(const float2*)(conf + e0);
    const float2* pp = (const float2*)(pts + (size_t)e0 * 3);  // 24B multiple -> 8B aligned
    float2 q0 = pp[0], q1 = pp[1], q2 = pp[2];
    float d0 = FAST_SQRT(q0.x * q0.x + q0.y * q0.y + q1.x * q1.x);
    float d1 = FAST_SQRT(q1.y * q1.y + q2.x * q2.x + q2.y * q2.y);
    float m0 = cc.x > 0.1f ? 1.0f : 0.0f;
    float m1 = cc.y > 0.1f ? 1.0f : 0.0f;
    v2f am  = {m0, m1};
    v2f ad  = {d0, d1};
    v2f bdm = {d0 * m0, d1 * m1};
#if HAVE_WMMA_F32
    c1 = __builtin_amdgcn_wmma_f32_16x16x4_f32(false, am, false, am,  (short)0, c1, false, false);
    c2 = __builtin_amdgcn_wmma_f32_16x16x4_f32(false, ad, false, am,  (short)0, c2, false, false);
    c3 = __builtin_amdgcn_wmma_f32_16x16x4_f32(false, ad, false, bdm, (short)0, c3, false, false);
#else
    c1[0] += m0 + m1;
    c2[0] += d0 * m0 + d1 * m1;
    c3[0] += d0 * d0 * m0 + d1 * d1 * m1;
#endif
  }
#if HAVE_WMMA_F32
  // diag(D): lanes 0..7 hold D[L][L] in vgpr L; lanes 24..31 hold D[L-16][L-16] in vgpr L-24
  int want = (L < 8) ? L : ((L >= 24) ? (L - 24) : -1);
  float t1 = 0.f, t2 = 0.f, t3 = 0.f;
#pragma unroll
  for (int r = 0; r < 8; ++r) {
    if (want == r) { t1 = c1[r]; t2 = c2[r]; t3 = c3[r]; }
  }
  if (want >= 0) {
    atomicAdd(&meta->sums[0], (double)t1);
    atomicAdd(&meta->sums[1], (double)t2);
    atomicAdd(&meta->sums[2], (double)t3);
  }
#else
  atomicAdd(&meta->sums[0], (double)c1[0]);
  atomicAdd(&meta->sums[1], (double)c2[0]);
  atomicAdd(&meta->sums[2], (double)c3[0]);
#endif
}

__global__ void k_fincomp(const Meta* meta, float* out) {
  double n = meta->sums[0], sd = meta->sums[1], sd2 = meta->sums[2];
  double mean = sd / n;
  double var = (sd2 - n * mean * mean) / (n - 1.0);
  double comp = sqrt(var) * (n / (double)NE);
  out[256] = (float)comp;
}

// ---------------- host launcher ----------------
extern "C" void kernel_launch(void* const* d_in, const int* in_sizes, int n_in,
                              void* d_out, int out_size, void* d_ws, size_t ws_size,
                              hipStream_t stream) {
  (void)in_sizes; (void)n_in; (void)out_size; (void)ws_size;
  const float* pts  = (const float*)d_in[0];   // [1,64,384,384,3]
  const float* conf = (const float*)d_in[1];   // [1,64,384,384]
  float* out = (float*)d_out;                  // 256 counts + 1 complexity
  char* ws = (char*)d_ws;
  unsigned* hist1 = (unsigned*)(ws + WS_HIST1);
  unsigned* hist2 = (unsigned*)(ws + WS_HIST2);
  Meta* meta = (Meta*)(ws + WS_META);

  k_init   <<<2048,  256, 0, stream>>>((unsigned*)ws, meta);
  k_hist1  <<<2048,  256, 0, stream>>>(conf, hist1);
  k_findbin<<<S,     256, 0, stream>>>(hist1, meta);
  k_hist2  <<<2048,  256, 0, stream>>>(conf, meta, hist2, meta);
  k_findmed<<<S,     256, 0, stream>>>(hist2, meta);
  k_bbox   <<<2048,  256, 0, stream>>>(pts, conf, meta);
  k_finbbox<<<1,     1,   0, stream>>>(meta);
  k_voxel  <<<4 * S, 256, 0, stream>>>(pts, conf, meta, out);
  k_complex<<<1024,  256, 0, stream>>>(pts, conf, meta);
  k_fincomp<<<1,     1,   0, stream>>>(meta, out);
}